// TreeNLLLoss_7627861917875
// MI455X (gfx1250) — compile-verified
//
#include <hip/hip_runtime.h>

// TreeNLLLoss for MI455X (gfx1250, wave32).
//
// Memory-bound gather + deterministic reduction:
//   per edge: idx loads (streamed) -> label gathers (L2-resident, 4MB)
//             -> unary gather (L2-resident, 64MB < 192MB L2)
//             -> edge_potentials gather (4B used per 1KB row; non-temporal,
//                ~128MB of random HBM cachelines => the dominant cost).
// One thread per edge maximizes memory-level parallelism for the 3-deep
// dependent-gather chain. Block partials are reduced with
// V_WMMA_F32_16X16X4_F32 (A = ones), then a single-block kernel finishes
// deterministically (no float atomics).

typedef __attribute__((ext_vector_type(2))) float v2f;
typedef __attribute__((ext_vector_type(8))) float v8f;

constexpr int kC = 16;         // num classes
constexpr int kBlock = 256;    // 8 wave32s per block

// Deterministic 256->1 block reduction using WMMA f32 16x16x4.
// All 256 threads must call this. Valid result only for threadIdx.x == 0.
__device__ __forceinline__ float block_reduce_wmma(float v, float* sm) {
    sm[threadIdx.x] = v;
    __syncthreads();
    float total = 0.0f;
    if (threadIdx.x < 32) {            // wave 0 only, EXEC = all 32 lanes
        const v2f a = {1.0f, 1.0f};    // A (16x4) = all ones across all lanes
        v8f c = {};                    // C = 0, accumulated over 4 WMMAs
#pragma unroll
        for (int j = 0; j < 4; ++j) {
            v2f b;                     // 64 partials -> B (4x16); slot placement
            b.x = sm[j * 64 + threadIdx.x];        // is irrelevant for the
            b.y = sm[j * 64 + 32 + threadIdx.x];   // grand total
            c = __builtin_amdgcn_wmma_f32_16x16x4_f32(
                    /*neg_a=*/false, a, /*neg_b=*/false, b,
                    /*c_mod=*/(short)0, c, /*reuse_a=*/false, /*reuse_b=*/false);
        }
        // A all-ones => every D row = the 16 column sums. c[0] on lanes 0..15
        // holds D[0, n]; sum of those 16 = sum of all 256 inputs.
        float t = c[0];
        t += __shfl_xor(t, 8, 32);
        t += __shfl_xor(t, 4, 32);
        t += __shfl_xor(t, 2, 32);
        t += __shfl_xor(t, 1, 32);
        total = t;                     // lane 0 of the 0..15 group has the sum
    }
    return total;
}

__global__ void tree_nll_edge_partials(
    const float* __restrict__ edge_pot,    // [E, 256]
    const float* __restrict__ unary,       // [N, 16]
    const int*   __restrict__ labels,      // [N]
    const int*   __restrict__ child_idx,   // [E]
    const int*   __restrict__ parent_idx,  // [E]
    float*       __restrict__ partials,    // [gridDim.x]
    int n_edges)
{
    __shared__ float sm[kBlock];
    float acc = 0.0f;
    const int stride = gridDim.x * blockDim.x;
    for (int i = blockIdx.x * blockDim.x + threadIdx.x; i < n_edges; i += stride) {
        const int cn = child_idx[i];
        const int pn = parent_idx[i];
        const int tc = labels[cn];
        const int tp = labels[pn];
        // Single-use random gather into the 1GB table: non-temporal so it
        // doesn't evict the L2-resident unary/label working set.
        const float e = __builtin_nontemporal_load(
            edge_pot + (long long)i * (kC * kC) + (kC * tc + tp));
        const float u = unary[(long long)cn * kC + tc];
        acc += e + u;
    }
    const float tot = block_reduce_wmma(acc, sm);
    if (threadIdx.x == 0) partials[blockIdx.x] = tot;
}

__global__ void tree_nll_finalize(
    const float* __restrict__ partials,
    int nb,
    const float* __restrict__ unary,
    const int*   __restrict__ labels,
    const int*   __restrict__ root_idx,
    const float* __restrict__ partition,
    float*       __restrict__ out)
{
    __shared__ float sm[kBlock];
    float acc = 0.0f;
    for (int i = threadIdx.x; i < nb; i += kBlock) acc += partials[i];
    const float tot = block_reduce_wmma(acc, sm);
    if (threadIdx.x == 0) {
        const int r = root_idx[0];
        const float loss = tot + unary[(long long)r * kC + labels[r]];
        out[0] = partition[0] - loss;   // -(loss - partition)
    }
}

extern "C" void kernel_launch(void* const* d_in, const int* in_sizes, int n_in,
                              void* d_out, int out_size, void* d_ws, size_t ws_size,
                              hipStream_t stream) {
    const float* edge_pot   = (const float*)d_in[0];  // [E, 256] f32
    const float* unary      = (const float*)d_in[1];  // [N, 16]  f32
    const int*   labels     = (const int*)  d_in[2];  // [N]      i32
    const int*   child_idx  = (const int*)  d_in[3];  // [E]      i32
    const int*   parent_idx = (const int*)  d_in[4];  // [E]      i32
    const int*   root_idx   = (const int*)  d_in[5];  // [1]      i32
    const float* partition  = (const float*)d_in[6];  // [1]      f32
    float*       out        = (float*)d_out;

    const int n_edges = in_sizes[3];                  // E

    int nb = (n_edges + kBlock - 1) / kBlock;         // one thread per edge
    const int ws_cap = (int)(ws_size / sizeof(float));
    if (nb > ws_cap) nb = ws_cap > 0 ? ws_cap : 1;
    if (nb < 1) nb = 1;

    float* partials = (float*)d_ws;

    tree_nll_edge_partials<<<nb, kBlock, 0, stream>>>(
        edge_pot, unary, labels, child_idx, parent_idx, partials, n_edges);

    tree_nll_finalize<<<1, kBlock, 0, stream>>>(
        partials, nb, unary, labels, root_idx, partition, out);
}